// AntiIoULoss_40312563040867
// MI455X (gfx1250) — compile-verified
//
#include <hip/hip_runtime.h>

// AntiIoULoss full-reduction for MI455X (gfx1250, wave32).
// Memory-bound: 176MB streamed once -> ~7.6us at 23.3 TB/s.
// V_WMMA_F32_16X16X4_F32 used as the intra-wave reduction engine.

typedef __attribute__((ext_vector_type(2))) float v2f;
typedef __attribute__((ext_vector_type(8))) float v8f;

#define C_CH   21
#define HW4    65536   // (512*512)/4 float4-groups per channel plane
#define NBLK   1024
#define NTHR   256
#define IGNORE 255.0f

// Reduce (pn, pd) across all 32 lanes of a wave using the f32 WMMA unit.
// A: VGPR0 = value, VGPR1 = 0  => A[m][0]=v_m, A[m][2]=v_{m+16}, A[m][1,3]=0
// B: all ones (4x16)           => D[m][n] = v_m + v_{m+16}  (every n)
// Sum D's 8 VGPRs in-lane: lanes 0-15 get rows 0..7, lanes 16-31 rows 8..15;
// one xor-16 shuffle completes the 32-lane sum (broadcast to all lanes).
__device__ __forceinline__ float2 wave_reduce_pair(float pn, float pd) {
    v2f an; an[0] = pn; an[1] = 0.0f;
    v2f ad; ad[0] = pd; ad[1] = 0.0f;
    v2f bo; bo[0] = 1.0f; bo[1] = 1.0f;
    v8f cz = {};
    v8f dn = __builtin_amdgcn_wmma_f32_16x16x4_f32(false, an, false, bo,
                                                   (short)0, cz, false, false);
    v8f dd = __builtin_amdgcn_wmma_f32_16x16x4_f32(false, ad, false, bo,
                                                   (short)0, cz, false, false);
    float tn = ((dn[0] + dn[1]) + (dn[2] + dn[3])) +
               ((dn[4] + dn[5]) + (dn[6] + dn[7]));
    float td = ((dd[0] + dd[1]) + (dd[2] + dd[3])) +
               ((dd[4] + dd[5]) + (dd[6] + dd[7]));
    tn += __shfl_xor(tn, 16, 32);
    td += __shfl_xor(td, 16, 32);
    return make_float2(tn, td);
}

__global__ void antiiou_stage1(const float* __restrict__ outputs,
                               float2* __restrict__ partials, int ngroups) {
    const float4* __restrict__ o4 = reinterpret_cast<const float4*>(outputs);
    float num_acc = 0.0f, den_acc = 0.0f;

    const int stride = gridDim.x * blockDim.x;
    for (int g = blockIdx.x * blockDim.x + threadIdx.x; g < ngroups; g += stride) {
        const int b = g >> 16;              // g / HW4  (512x512 plane)
        const int r = g & (HW4 - 1);        // float4 index inside plane
        const size_t base = (size_t)b * ((size_t)C_CH * HW4) + (size_t)r;

        float4 O = make_float4(0.f, 0.f, 0.f, 0.f);
        float4 S = O, T = O, Q = O;
#pragma unroll
        for (int c = 0; c < C_CH; ++c) {
            float4 o = o4[base + (size_t)c * HW4];
            float m, om;
            m = (o.x != IGNORE) ? 1.0f : 0.0f; om = o.x * m;
            O.x += o.x; T.x += m; S.x += om; Q.x = fmaf(o.x, om, Q.x);
            m = (o.y != IGNORE) ? 1.0f : 0.0f; om = o.y * m;
            O.y += o.y; T.y += m; S.y += om; Q.y = fmaf(o.y, om, Q.y);
            m = (o.z != IGNORE) ? 1.0f : 0.0f; om = o.z * m;
            O.z += o.z; T.z += m; S.z += om; Q.z = fmaf(o.z, om, Q.z);
            m = (o.w != IGNORE) ? 1.0f : 0.0f; om = o.w * m;
            O.w += o.w; T.w += m; S.w += om; Q.w = fmaf(o.w, om, Q.w);
        }
        // num_pix = O*S - Q ; den_contrib_pix = O*T + 19*S
        num_acc += (fmaf(O.x, S.x, -Q.x) + fmaf(O.y, S.y, -Q.y)) +
                   (fmaf(O.z, S.z, -Q.z) + fmaf(O.w, S.w, -Q.w));
        den_acc += (fmaf(O.x, T.x, 19.0f * S.x) + fmaf(O.y, T.y, 19.0f * S.y)) +
                   (fmaf(O.z, T.z, 19.0f * S.z) + fmaf(O.w, T.w, 19.0f * S.w));
    }

    float2 wr = wave_reduce_pair(num_acc, den_acc);

    __shared__ float2 sp[NTHR / 32];
    const int wave = threadIdx.x >> 5;
    if ((threadIdx.x & 31) == 0) sp[wave] = wr;
    __syncthreads();
    if (threadIdx.x == 0) {
        float n = 0.0f, d = 0.0f;
        const int nw = blockDim.x >> 5;
        for (int i = 0; i < nw; ++i) { n += sp[i].x; d += sp[i].y; }
        partials[blockIdx.x] = make_float2(n, d);
    }
}

__global__ void antiiou_stage2(const float2* __restrict__ partials, int nparts,
                               float* __restrict__ out) {
    float n = 0.0f, d = 0.0f;
    for (int i = threadIdx.x; i < nparts; i += blockDim.x) {
        float2 p = partials[i];
        n += p.x; d += p.y;
    }
    float2 wr = wave_reduce_pair(n, d);

    __shared__ float2 sp[NTHR / 32];
    const int wave = threadIdx.x >> 5;
    if ((threadIdx.x & 31) == 0) sp[wave] = wr;
    __syncthreads();
    if (threadIdx.x == 0) {
        float num = 0.0f, denp = 0.0f;
        const int nw = blockDim.x >> 5;
        for (int i = 0; i < nw; ++i) { num += sp[i].x; denp += sp[i].y; }
        const float den = denp - num;   // den = den_a + den_b - num
        out[0] = num / den;
    }
}

extern "C" void kernel_launch(void* const* d_in, const int* in_sizes, int n_in,
                              void* d_out, int out_size, void* d_ws, size_t ws_size,
                              hipStream_t stream) {
    const float* outputs = (const float*)d_in[0];   // [8,21,512,512] f32
    // d_in[1] (targets) is ignored by the reference math.
    float* out = (float*)d_out;                     // single scalar
    float2* partials = (float2*)d_ws;

    const int ngroups = in_sizes[0] / (C_CH * 4);   // float4 pixel-groups
    int nblk = NBLK;
    const int max_blk = (int)(ws_size / sizeof(float2));
    if (nblk > max_blk) nblk = max_blk;
    if (nblk < 1) nblk = 1;

    antiiou_stage1<<<nblk, NTHR, 0, stream>>>(outputs, partials, ngroups);
    antiiou_stage2<<<1, NTHR, 0, stream>>>(partials, nblk, out);
}